// ObliviousDecisionTree_2027224564349
// MI455X (gfx1250) — compile-verified
//
#include <hip/hip_runtime.h>

typedef __attribute__((ext_vector_type(2))) float v2f;
typedef __attribute__((ext_vector_type(8))) float v8f;

#define BATCH   8192
#define DIM     128
#define NTREES  512
#define TDEPTH  6
#define UNITS   8
#define NLEAF   64
#define NCOL    (NTREES * TDEPTH)   // 3072

#define TREES_PER_OCT   8           // 8 trees * 6 levels = 48 cols = 3 WMMA N-tiles
#define OCTS_PER_BLOCK  8           // 64 trees per block
#define TREES_PER_BLOCK (TREES_PER_OCT * OCTS_PER_BLOCK)
#define NGROUPS         (NTREES / TREES_PER_BLOCK)     // 8
#define WAVES_PER_BLOCK 4
#define ROWS_PER_WAVE   16
#define ROWS_PER_BLOCK  (WAVES_PER_BLOCK * ROWS_PER_WAVE)  // 64

// ---------------------------------------------------------------------------
// Kernel 1: sparsemax along dim for each of 3072 (tree,level) rows.
// fsl layout: (dim, tree, level) -> element (i, r) at fsl[i*NCOL + r].
// Output sel keeps the same (k-major) layout, which is exactly the B-matrix
// layout GEMM1 wants: sel[k*NCOL + col].
// ---------------------------------------------------------------------------
__global__ __launch_bounds__(256) void odt_sparsemax_kernel(
    const float* __restrict__ fsl, float* __restrict__ sel)
{
    __shared__ float tile[DIM][33];   // padded stride: conflict-free both ways
    const int tid = threadIdx.x;
    const int r0  = blockIdx.x * 32;  // 32 rows per block

    // coalesced load: consecutive threads read consecutive r for fixed i
    #pragma unroll
    for (int chunk = 0; chunk < DIM / 8; ++chunk) {
        int i  = chunk * 8 + (tid >> 5);
        int rr = tid & 31;
        tile[i][rr] = fsl[i * NCOL + r0 + rr];
    }
    __syncthreads();

    const int wave = tid >> 5;
    const int lane = tid & 31;

    // each wave handles 4 rows; lane holds 4 of the 128 features
    for (int rr = wave * 4; rr < wave * 4 + 4; ++rr) {
        float z[4];
        #pragma unroll
        for (int j = 0; j < 4; ++j) z[j] = tile[lane + 32 * j][rr];

        float m = fmaxf(fmaxf(z[0], z[1]), fmaxf(z[2], z[3]));
        #pragma unroll
        for (int off = 16; off; off >>= 1) m = fmaxf(m, __shfl_xor(m, off, 32));

        // bisection on tau: s(tau) = sum(relu(z - tau)) is decreasing;
        // s(m-1) >= 1, s(m) = 0.
        float lo = m - 1.0f, hi = m;
        for (int it = 0; it < 32; ++it) {
            float mid = 0.5f * (lo + hi);
            float s = 0.0f;
            #pragma unroll
            for (int j = 0; j < 4; ++j) s += fmaxf(z[j] - mid, 0.0f);
            #pragma unroll
            for (int off = 16; off; off >>= 1) s += __shfl_xor(s, off, 32);
            if (s >= 1.0f) lo = mid; else hi = mid;
        }
        // exact refinement once support set is identified
        float tau0 = 0.5f * (lo + hi);
        float cnt = 0.0f, ssum = 0.0f;
        #pragma unroll
        for (int j = 0; j < 4; ++j) {
            if (z[j] > tau0) { cnt += 1.0f; ssum += z[j]; }
        }
        #pragma unroll
        for (int off = 16; off; off >>= 1) {
            cnt  += __shfl_xor(cnt,  off, 32);
            ssum += __shfl_xor(ssum, off, 32);
        }
        float tau = (ssum - 1.0f) / fmaxf(cnt, 1.0f);
        #pragma unroll
        for (int j = 0; j < 4; ++j)
            tile[lane + 32 * j][rr] = fmaxf(z[j] - tau, 0.0f);  // own column only
    }
    __syncthreads();

    #pragma unroll
    for (int chunk = 0; chunk < DIM / 8; ++chunk) {
        int i  = chunk * 8 + (tid >> 5);
        int rr = tid & 31;
        sel[i * NCOL + r0 + rr] = tile[i][rr];
    }
}

// ---------------------------------------------------------------------------
// Kernel 2: fused fv-GEMM + gating + leaf aggregation + response-GEMM.
// Block: 4 waves x 16 batch rows = 64 rows; 64 trees (one tree-group).
// All matrix math on V_WMMA_F32_16X16X4_F32 (full fp32 precision).
// ---------------------------------------------------------------------------
__global__ __launch_bounds__(128) void odt_main_kernel(
    const float* __restrict__ inputs,   // (8192, 128)
    const float* __restrict__ sel,      // (128, 3072) k-major
    const float* __restrict__ thr,      // (512, 6)
    const float* __restrict__ logt,     // (512, 6)
    const float* __restrict__ resp,     // (512, 8, 64)
    float* __restrict__ partial)        // (NGROUPS, 8192, 8)
{
    __shared__ float gates[WAVES_PER_BLOCK][ROWS_PER_WAVE][TREES_PER_OCT][TDEPTH][2];

    const int tid  = threadIdx.x;
    const int wave = tid >> 5;
    const int lane = tid & 31;
    const int lo   = lane & 15;     // M / N index inside 16x16 tile
    const int hi   = lane >> 4;     // half-wave select
    const int row0 = blockIdx.x * ROWS_PER_BLOCK + wave * ROWS_PER_WAVE;
    const int grp  = blockIdx.y;

    v8f acc_out = {};   // out[row=v+8*hi][unit=lo], summed over all 64 trees

    for (int oct = 0; oct < OCTS_PER_BLOCK; ++oct) {
        const int t0   = grp * TREES_PER_BLOCK + oct * TREES_PER_OCT;
        const int col0 = t0 * TDEPTH;            // 48 columns for this octet

        // ---- GEMM1: fv(16 x 48) = inputs(16 x 128) x sel(128 x 48) ----
        v8f fv[3];
        fv[0] = (v8f){}; fv[1] = (v8f){}; fv[2] = (v8f){};
        #pragma unroll 4
        for (int kk = 0; kk < DIM; kk += 4) {
            const int ka = kk + 2 * hi;          // A VGPRj <-> K = 2*hi + j
            v2f a = *(const v2f*)(inputs + (size_t)(row0 + lo) * DIM + ka);
            #pragma unroll
            for (int t = 0; t < 3; ++t) {
                const int col = col0 + 16 * t + lo;
                v2f b;                           // B VGPRv <-> K = 2*hi + v
                b.x = sel[(size_t)(ka    ) * NCOL + col];
                b.y = sel[(size_t)(ka + 1) * NCOL + col];
                fv[t] = __builtin_amdgcn_wmma_f32_16x16x4_f32(
                            false, a, false, b, (short)0, fv[t], false, false);
            }
        }

        // ---- gating: sparsemoid(+/-(fv - thr) * exp(-logT)) -> LDS ----
        #pragma unroll
        for (int t = 0; t < 3; ++t) {
            const int col = 16 * t + lo;         // 0..47 within octet
            const int tt  = col / TDEPTH;
            const int lvl = col - tt * TDEPTH;
            const int tg  = t0 + tt;
            const float th   = thr[tg * TDEPTH + lvl];
            const float itmp = expf(-logt[tg * TDEPTH + lvl]);
            #pragma unroll
            for (int v = 0; v < 8; ++v) {
                const float tl = (fv[t][v] - th) * itmp;
                const float gp = fminf(fmaxf(0.5f * tl + 0.5f, 0.0f), 1.0f);
                const float gn = fminf(fmaxf(0.5f - 0.5f * tl, 0.0f), 1.0f);
                const int row = v + 8 * hi;
                gates[wave][row][tt][lvl][0] = gn;   // LUT bit==1 branch
                gates[wave][row][tt][lvl][1] = gp;   // LUT bit==0 branch
            }
        }
        __syncthreads();

        // ---- GEMM2: out += agg(16 x 64) x resp_t(64 x 16pad), per tree ----
        // Leaf index decomposition for this lane's A-fragment values:
        //   c = kk + 2*hi + j  ->  bit0 = j (compile-time), bit1 = hi,
        //   bits 2..5 = kk>>2 (compile-time after unroll).
        // Build pair-products once per tree, then 2 muls per leaf value.
        #pragma unroll 1
        for (int tt = 0; tt < TREES_PER_OCT; ++tt) {
            const int tg = t0 + tt;
            if (tg + 1 < NTREES)   // hide L2 latency of next tree's response
                __builtin_prefetch(resp + (size_t)(tg + 1) * UNITS * NLEAF, 0, 0);

            const float gn0 = gates[wave][lo][tt][0][0], gp0 = gates[wave][lo][tt][0][1];
            const float gn1 = gates[wave][lo][tt][1][0], gp1 = gates[wave][lo][tt][1][1];
            const float gn2 = gates[wave][lo][tt][2][0], gp2 = gates[wave][lo][tt][2][1];
            const float gn3 = gates[wave][lo][tt][3][0], gp3 = gates[wave][lo][tt][3][1];
            const float gn4 = gates[wave][lo][tt][4][0], gp4 = gates[wave][lo][tt][4][1];
            const float gn5 = gates[wave][lo][tt][5][0], gp5 = gates[wave][lo][tt][5][1];

            const float q1  = hi ? gn1 : gp1;    // bit1 select, once per tree
            const float axb = gp0 * q1;          // a.x: bit0 = 0
            const float ayb = gn0 * q1;          // a.y: bit0 = 1
            const float s23[4] = { gp2 * gp3, gn2 * gp3, gp2 * gn3, gn2 * gn3 };
            const float s45[4] = { gp4 * gp5, gn4 * gp5, gp4 * gn5, gn4 * gn5 };

            const float* rbase = resp + (size_t)tg * UNITS * NLEAF;
            #pragma unroll
            for (int kk = 0; kk < NLEAF; kk += 4) {
                const int u = kk >> 2;           // 0..15, compile-time
                const float ph = s23[u & 3] * s45[u >> 2];
                v2f a;
                a.x = ph * axb;
                a.y = ph * ayb;
                v2f b = {0.0f, 0.0f};            // N = lo; units 8..15 padded
                if (lo < UNITS) b = *(const v2f*)(rbase + lo * NLEAF + kk + 2 * hi);
                acc_out = __builtin_amdgcn_wmma_f32_16x16x4_f32(
                              false, a, false, b, (short)0, acc_out, false, false);
            }
        }
        __syncthreads();   // gates reused next octet
    }

    // C layout: VGPR v, lane -> row = v + 8*hi, unit = lo
    if (lo < UNITS) {
        #pragma unroll
        for (int v = 0; v < 8; ++v) {
            const int row = row0 + v + 8 * hi;
            partial[((size_t)grp * BATCH + row) * UNITS + lo] = acc_out[v];
        }
    }
}

// ---------------------------------------------------------------------------
// Kernel 3: deterministic reduction over tree groups + mean scale.
// ---------------------------------------------------------------------------
__global__ __launch_bounds__(256) void odt_reduce_kernel(
    const float* __restrict__ partial, float* __restrict__ out)
{
    const int idx = blockIdx.x * blockDim.x + threadIdx.x;
    if (idx < BATCH * UNITS) {
        float s = 0.0f;
        #pragma unroll
        for (int g = 0; g < NGROUPS; ++g)
            s += partial[(size_t)g * BATCH * UNITS + idx];
        out[idx] = s * (1.0f / NTREES);
    }
}

extern "C" void kernel_launch(void* const* d_in, const int* in_sizes, int n_in,
                              void* d_out, int out_size, void* d_ws, size_t ws_size,
                              hipStream_t stream)
{
    (void)in_sizes; (void)n_in; (void)out_size; (void)ws_size;
    const float* inputs = (const float*)d_in[0];   // (8192, 128)
    const float* fsl    = (const float*)d_in[1];   // (128, 512, 6)
    const float* thr    = (const float*)d_in[2];   // (512, 6)
    const float* logt   = (const float*)d_in[3];   // (512, 6)
    const float* resp   = (const float*)d_in[4];   // (512, 8, 64)
    float* out = (float*)d_out;                    // (8192, 8)

    char* ws = (char*)d_ws;
    float* sel     = (float*)ws;                                   // 128*3072*4 = 1.5 MB
    float* partial = (float*)(ws + (size_t)DIM * NCOL * sizeof(float)); // 8*8192*8*4 = 2 MB

    odt_sparsemax_kernel<<<NCOL / 32, 256, 0, stream>>>(fsl, sel);

    dim3 grid(BATCH / ROWS_PER_BLOCK, NGROUPS);    // (128, 8)
    odt_main_kernel<<<grid, 128, 0, stream>>>(inputs, sel, thr, logt, resp, partial);

    odt_reduce_kernel<<<(BATCH * UNITS + 255) / 256, 256, 0, stream>>>(partial, out);
}